// ExpSSGL_Encoder_48000554500967
// MI455X (gfx1250) — compile-verified
//
#include <hip/hip_runtime.h>
#include <hip/hip_bf16.h>
#include <stdint.h>

#define EMB   64
#define TILE  256   // edges staged per block (256 threads = 8 waves, wave32)

// ---------------------------------------------------------------------------
// CDNA5 async global->LDS staging (ASYNCcnt path). Guarded so a missing
// builtin degrades to a plain LDS store instead of breaking the build.
// ---------------------------------------------------------------------------
#if defined(__HIP_DEVICE_COMPILE__) && defined(__gfx1250__) && \
    __has_builtin(__builtin_amdgcn_global_load_async_to_lds_b32)
#define HAVE_ASYNC_LDS 1
#else
#define HAVE_ASYNC_LDS 0
#endif

typedef __attribute__((address_space(1))) int g_i32_t;   // global int
typedef __attribute__((address_space(3))) int l_i32_t;   // LDS int

__device__ __forceinline__ void stage_b32(void* lds_dst, const void* g_src) {
#if HAVE_ASYNC_LDS
  __builtin_amdgcn_global_load_async_to_lds_b32(
      (g_i32_t*)(g_src), (l_i32_t*)(lds_dst), 0, 0);
#else
  *(uint32_t*)lds_dst = *(const uint32_t*)g_src;
#endif
}

__device__ __forceinline__ void stage_wait() {
#if HAVE_ASYNC_LDS
  asm volatile("s_wait_asynccnt 0" ::: "memory");
#endif
}

// ---------------------------------------------------------------------------
// COO SpMM scatter: y[row] += val * x[col], cooperative 32-lane per edge.
// x is split across (x0 with rows [0,split)) and (x1 with rows [split,...));
// pass split = N, x0 == x1 for a single dense source.
// ---------------------------------------------------------------------------
__global__ void spmm_scatter(const float* __restrict__ x0,
                             const float* __restrict__ x1,
                             int split,
                             const float* __restrict__ vals,
                             const int*   __restrict__ rows,
                             const int*   __restrict__ cols,
                             float* __restrict__ y,
                             int nnz) {
  __shared__ int   s_rows[TILE];
  __shared__ int   s_cols[TILE];
  __shared__ float s_vals[TILE];

  const int t    = threadIdx.x;
  const int base = blockIdx.x * TILE;
  const int g    = base + t;

  if (g < nnz) {
    stage_b32(&s_rows[t], &rows[g]);
    stage_b32(&s_cols[t], &cols[g]);
    stage_b32(&s_vals[t], &vals[g]);
  }
  stage_wait();        // this wave's async copies are in LDS
  __syncthreads();     // all waves' copies are in LDS

  const int wave = t >> 5;
  const int lane = t & 31;
  const int e0   = wave << 5;

  for (int i = 0; i < 32; ++i) {
    const int e = base + e0 + i;
    if (e >= nnz) break;
    const int   r = s_rows[e0 + i];   // uniform across wave -> broadcast read
    const int   c = s_cols[e0 + i];
    const float v = s_vals[e0 + i];

    const float* src = (c < split) ? (x0 + (size_t)c * EMB)
                                   : (x1 + (size_t)(c - split) * EMB);
    // 32 lanes x float2 = one coalesced 256B row gather (global_load_b64/lane)
    const float2 xv = *(const float2*)(src + lane * 2);

    float* dst = y + (size_t)r * EMB + lane * 2;
    unsafeAtomicAdd(dst + 0, v * xv.x);   // global_atomic_add_f32
    unsafeAtomicAdd(dst + 1, v * xv.y);
  }
}

// ---------------------------------------------------------------------------
// Elementwise helpers (float4, L2-bandwidth bound, negligible vs SpMM)
// ---------------------------------------------------------------------------
__global__ void zero_f4(float4* __restrict__ p, int n4) {
  int i = blockIdx.x * blockDim.x + threadIdx.x;
  if (i < n4) p[i] = make_float4(0.f, 0.f, 0.f, 0.f);
}

__global__ void acc_add_f4(float4* __restrict__ acc, const float4* __restrict__ x,
                           int n4) {
  int i = blockIdx.x * blockDim.x + threadIdx.x;
  if (i < n4) {
    float4 a = acc[i], b = x[i];
    a.x += b.x; a.y += b.y; a.z += b.z; a.w += b.w;
    acc[i] = a;
  }
}

__global__ void final_f4(float4* __restrict__ acc, const float4* __restrict__ x,
                         float s, int n4) {
  int i = blockIdx.x * blockDim.x + threadIdx.x;
  if (i < n4) {
    float4 a = acc[i], b = x[i];
    a.x = (a.x + b.x) * s; a.y = (a.y + b.y) * s;
    a.z = (a.z + b.z) * s; a.w = (a.w + b.w) * s;
    acc[i] = a;
  }
}

// ---------------------------------------------------------------------------
// Orchestration: acc = (A1 e + A2 e + A3 e) / 3 with ping-pong SpMM buffers
// ---------------------------------------------------------------------------
extern "C" void kernel_launch(void* const* d_in, const int* in_sizes, int n_in,
                              void* d_out, int out_size, void* d_ws, size_t ws_size,
                              hipStream_t stream) {
  const float* user_emb = (const float*)d_in[0];
  const float* item_emb = (const float*)d_in[1];
  const float* vals     = (const float*)d_in[2];
  const int*   rows     = (const int*)d_in[3];
  const int*   cols     = (const int*)d_in[4];

  const int user_n = in_sizes[0] / EMB;      // 80000
  const int item_n = in_sizes[1] / EMB;      // 40000
  const int N      = user_n + item_n;        // 120000
  const int nnz    = in_sizes[2];            // 1.2M
  const size_t NE  = (size_t)N * EMB;        // 7.68M floats

  float* A   = (float*)d_ws;        // layer output ping
  float* B   = A + NE;              // layer output pong
  float* acc = (float*)d_out;       // accumulator == final output layout

  const int n4 = (int)(NE / 4);
  const dim3 eb(256), eg((n4 + 255) / 256);
  const dim3 sb(TILE), sg((nnz + TILE - 1) / TILE);

  // zero A, B, acc
  zero_f4<<<eg, eb, 0, stream>>>((float4*)A,   n4);
  zero_f4<<<eg, eb, 0, stream>>>((float4*)B,   n4);
  zero_f4<<<eg, eb, 0, stream>>>((float4*)acc, n4);

  // layer 1: e1 = A @ concat(user_emb, item_emb)  -> A ; acc += e1
  spmm_scatter<<<sg, sb, 0, stream>>>(user_emb, item_emb, user_n,
                                      vals, rows, cols, A, nnz);
  acc_add_f4<<<eg, eb, 0, stream>>>((float4*)acc, (const float4*)A, n4);

  // layer 2: e2 = A @ e1 -> B ; acc += e2
  spmm_scatter<<<sg, sb, 0, stream>>>(A, A, N, vals, rows, cols, B, nnz);
  acc_add_f4<<<eg, eb, 0, stream>>>((float4*)acc, (const float4*)B, n4);

  // layer 3: e3 = A @ e2 -> A (re-zeroed) ; acc = (acc + e3) / 3
  zero_f4<<<eg, eb, 0, stream>>>((float4*)A, n4);
  spmm_scatter<<<sg, sb, 0, stream>>>(B, B, N, vals, rows, cols, A, nnz);
  final_f4<<<eg, eb, 0, stream>>>((float4*)acc, (const float4*)A,
                                  1.0f / 3.0f, n4);
}